// DepthNet_20512763806141
// MI455X (gfx1250) — compile-verified
//
#include <hip/hip_runtime.h>
#include <cstdint>

// Problem constants (from reference): B,V,C,H,W,D = 2,3,8,128,160,64
#define B_  2
#define V_  3
#define C_  8
#define H_  128
#define W_  160
#define D_  64
#define HW_ (H_ * W_)        // 20480

#define PIX_PER_BLOCK 32     // pixels per block
#define CHUNKS        4      // depth chunks per pixel
#define DCHUNK        (D_ / CHUNKS)   // 16 depths per chunk

typedef __attribute__((ext_vector_type(2))) float v2f;
typedef __attribute__((ext_vector_type(8))) float v8f;
// unaligned (4B) pair load type for corner-pair gathers
typedef float v2fu __attribute__((ext_vector_type(2), aligned(4)));

#if __has_builtin(__builtin_amdgcn_wmma_f32_16x16x4_f32)
#define USE_WMMA 1
#else
#define USE_WMMA 0
#endif

__global__ __launch_bounds__(256)
void mvs_depth_kernel(const float* __restrict__ feat,   // [V,B,C,H,W]
                      const float* __restrict__ proj,   // [B,V,3,4]
                      const float* __restrict__ dvals,  // [B,D,H,W]
                      const float* __restrict__ regw,   // [C]
                      float* __restrict__ out)          // [B,H,W]
{
    __shared__ float lds_dv[D_ * PIX_PER_BLOCK];   // 8 KB: depth_values tile
    __shared__ float pm[CHUNKS][PIX_PER_BLOCK];    // per-chunk softmax max
    __shared__ float ps[CHUNKS][PIX_PER_BLOCK];    // per-chunk sum(exp)
    __shared__ float pt[CHUNKS][PIX_PER_BLOCK];    // per-chunk sum(dv*exp)

    const int tid   = threadIdx.x;
    const int lane  = tid & 31;
    const int wv    = tid >> 5;        // wave 0..7
    const int grp   = wv & 1;          // pixel group (16 pixels each)
    const int chunk = wv >> 1;         // depth chunk 0..3
    const int half  = lane >> 4;       // lane pair half: 0 -> ch{0,1,4,5}, 1 -> ch{2,3,6,7}
    const int m16   = lane & 15;       // pixel row M within wave tile
    const int pix   = grp * 16 + m16;  // pixel within block, 0..31
    const int P     = blockIdx.x * PIX_PER_BLOCK + pix;
    const int b     = P / HW_;
    const int rem   = P - b * HW_;
    const int y     = rem / W_;
    const int x     = rem - y * W_;

    // ---- async stage of depth_values tile into LDS (ASYNCcnt path) ----
    {
        const int spix  = tid & 31;
        const int dslot = tid >> 5;                 // 0..7, each thread loads 8 depths
        const int SP    = blockIdx.x * PIX_PER_BLOCK + spix;
        const int sb    = SP / HW_;
        const int srem  = SP - sb * HW_;
#pragma unroll
        for (int j = 0; j < 8; ++j) {
            const int d = dslot * 8 + j;
            const float* gsrc = dvals + (size_t)(sb * D_ + d) * HW_ + srem;
            const uint32_t ldst = (uint32_t)(uintptr_t)(&lds_dv[d * PIX_PER_BLOCK + spix]);
            const uint64_t gad  = (uint64_t)(uintptr_t)gsrc;
            asm volatile("global_load_async_to_lds_b32 %0, %1, off"
                         :: "v"(ldst), "v"(gad) : "memory");
        }
    }

    // ---- overlap: per-pixel projection setup while the DMA is in flight ----
    float Rp[2][3], Tv[2][3];
#pragma unroll
    for (int vv = 0; vv < 2; ++vv) {
        const float* pmx = proj + (size_t)((b * V_ + (vv + 1)) * 3) * 4;
#pragma unroll
        for (int r = 0; r < 3; ++r) {
            const float r0 = pmx[r * 4 + 0], r1 = pmx[r * 4 + 1];
            const float r2 = pmx[r * 4 + 2], r3 = pmx[r * 4 + 3];
            Rp[vv][r] = r0 * (float)x + r1 * (float)y + r2;  // R @ [x,y,1]
            Tv[vv][r] = r3;
        }
    }

    // lane-pair channel split matching WMMA A-matrix (16x4 f32) layout
    const int ch0 = half ? 2 : 0;
    const int chs[4] = { ch0, ch0 + 1, ch0 + 4, ch0 + 5 };

    // reference-view features for my 4 channels (depth-invariant)
    const float* fb0 = feat + (size_t)(b * C_) * HW_ + (size_t)y * W_ + x;
    float refv[4];
#pragma unroll
    for (int k = 0; k < 4; ++k) refv[k] = fb0[(size_t)chs[k] * HW_];

    // reg_weight, arranged as the B-matrix rows for the two WMMAs
    float w8[8];
#pragma unroll
    for (int i = 0; i < 8; ++i) w8[i] = regw[i];
    const float bA0 = half ? w8[2] : w8[0];   // == w8[chs[0]]
    const float bA1 = half ? w8[3] : w8[1];   // == w8[chs[1]]
    const float bB0 = half ? w8[6] : w8[4];   // == w8[chs[2]]
    const float bB1 = half ? w8[7] : w8[5];   // == w8[chs[3]]

    const float* fbase[2];
    fbase[0] = feat + (size_t)((1 * B_ + b) * C_) * HW_;
    fbase[1] = feat + (size_t)((2 * B_ + b) * C_) * HW_;

    asm volatile("s_wait_asynccnt 0" ::: "memory");
    __syncthreads();   // LDS depth tile ready for all waves

    // online-softmax state (WMMA path keeps 8 sets in the transposed D layout)
    float sm[8], ss[8], st[8];
#pragma unroll
    for (int j = 0; j < 8; ++j) { sm[j] = -3.402823466e38f; ss[j] = 0.f; st[j] = 0.f; }

#pragma unroll 2
    for (int dd = 0; dd < DCHUNK; ++dd) {
        const int d = chunk * DCHUNK + dd;
        const float dv   = lds_dv[d * PIX_PER_BLOCK + pix];
        const float invd = __builtin_amdgcn_rcpf(dv);

        float sum[4], sq[4];
#pragma unroll
        for (int k = 0; k < 4; ++k) { sum[k] = refv[k]; sq[k] = refv[k] * refv[k]; }
        float cnt = 1.0f;

#pragma unroll
        for (int vv = 0; vv < 2; ++vv) {
            const float s0 = fmaf(Tv[vv][0], invd, Rp[vv][0]);
            const float s1 = fmaf(Tv[vv][1], invd, Rp[vv][1]);
            const float s2 = fmaf(Tv[vv][2], invd, Rp[vv][2]);
            const float invz = __builtin_amdgcn_rcpf(s2);
            const float sx = s0 * invz;     // == ix of reference
            const float sy = s1 * invz;     // == iy of reference
            if (sx > 0.f && sx < (float)(W_ - 1) && sy > 0.f && sy < (float)(H_ - 1))
                cnt += 1.0f;

            const float fx = floorf(sx), fy = floorf(sy);
            const float wx1 = sx - fx, wy1 = sy - fy;
            const float wx0 = 1.f - wx1, wy0 = 1.f - wy1;
            // per-corner validity (zeros padding, matches reference masks)
            const float vx0 = (fx >=  0.f && fx <= (float)(W_ - 1)) ? 1.f : 0.f;
            const float vx1 = (fx >= -1.f && fx <= (float)(W_ - 2)) ? 1.f : 0.f;
            const float vy0 = (fy >=  0.f && fy <= (float)(H_ - 1)) ? 1.f : 0.f;
            const float vy1 = (fy >= -1.f && fy <= (float)(H_ - 2)) ? 1.f : 0.f;
            const float w00 = wx0 * wy0 * vx0 * vy0;
            const float w10 = wx1 * wy0 * vx1 * vy0;
            const float w01 = wx0 * wy1 * vx0 * vy1;
            const float w11 = wx1 * wy1 * vx1 * vy1;

            // clamped coords; pair-load anchor kept in [0, W-2] so the b64
            // never crosses the tensor end; selects recover clamp semantics
            const int ax0 = (int)fminf(fmaxf(fx, 0.f), (float)(W_ - 2));
            const int cy0 = (int)fminf(fmaxf(fy,       0.f), (float)(H_ - 1));
            const int cy1 = (int)fminf(fmaxf(fy + 1.f, 0.f), (float)(H_ - 1));
            const int cx0 = (int)fminf(fmaxf(fx,       0.f), (float)(W_ - 1));
            const int cx1 = (int)fminf(fmaxf(fx + 1.f, 0.f), (float)(W_ - 1));
            const bool sel0 = (cx0 == ax0);
            const bool sel1 = (cx1 == ax0);
            const int ro0 = cy0 * W_ + ax0;
            const int ro1 = cy1 * W_ + ax0;
            const float* fb = fbase[vv];

#pragma unroll
            for (int k = 0; k < 4; ++k) {
                const float* cb = fb + (size_t)chs[k] * HW_;
                const v2fu r0 = *(const v2fu*)(cb + ro0);  // corners (x0,y0),(x1,y0)
                const v2fu r1 = *(const v2fu*)(cb + ro1);  // corners (x0,y1),(x1,y1)
                const float v00 = sel0 ? r0.x : r0.y;
                const float v10 = sel1 ? r0.x : r0.y;
                const float v01 = sel0 ? r1.x : r1.y;
                const float v11 = sel1 ? r1.x : r1.y;
                float wp = w00 * v00;
                wp = fmaf(w10, v10, wp);
                wp = fmaf(w01, v01, wp);
                wp = fmaf(w11, v11, wp);
                sum[k] += wp;
                sq[k]   = fmaf(wp, wp, sq[k]);
            }
        }

        const float cinv = __builtin_amdgcn_rcpf(cnt);
        float var_[4];
#pragma unroll
        for (int k = 0; k < 4; ++k) {
            const float mu = sum[k] * cinv;
            var_[k] = fmaf(-mu, mu, sq[k] * cinv);
        }

#if USE_WMMA
        // cost[m] = sum_c w_c * var_c[m], two chained V_WMMA_F32_16X16X4_F32
        v2f A1 = { var_[0], var_[1] };
        v2f A2 = { var_[2], var_[3] };
        v2f B1 = { bA0, bA1 };
        v2f B2 = { bB0, bB1 };
        v8f acc = {};
        acc = __builtin_amdgcn_wmma_f32_16x16x4_f32(false, A1, false, B1,
                                                    (short)0, acc, false, false);
        acc = __builtin_amdgcn_wmma_f32_16x16x4_f32(false, A2, false, B2,
                                                    (short)0, acc, false, false);
        // online softmax in the transposed D layout: set j tracks pixel
        // grp*16 + j + 8*half; only lanes j and 24+j hold the matching dv.
#pragma unroll
        for (int j = 0; j < 8; ++j) {
            const float cst = acc[j];
            const float nm  = fmaxf(sm[j], cst);
            const float e1  = __expf(sm[j] - nm);
            const float e2  = __expf(cst  - nm);
            ss[j] = fmaf(ss[j], e1, e2);
            st[j] = fmaf(st[j], e1, dv * e2);
            sm[j] = nm;
        }
#else
        // VALU fallback: partner-lane reduction across the channel split
        float part = var_[0] * bA0;
        part = fmaf(var_[1], bA1, part);
        part = fmaf(var_[2], bB0, part);
        part = fmaf(var_[3], bB1, part);
        const float cst = part + __shfl_xor(part, 16, 32);
        const float nm  = fmaxf(sm[0], cst);
        const float e1  = __expf(sm[0] - nm);
        const float e2  = __expf(cst  - nm);
        ss[0] = fmaf(ss[0], e1, e2);
        st[0] = fmaf(st[0], e1, dv * e2);
        sm[0] = nm;
#endif
    }

    // ---- per-chunk partials to LDS ----
#if USE_WMMA
#pragma unroll
    for (int j = 0; j < 8; ++j) {
        if (lane == (unsigned)j || lane == (unsigned)(24 + j)) {
            const int pp = grp * 16 + j + (half ? 8 : 0);
            pm[chunk][pp] = sm[j];
            ps[chunk][pp] = ss[j];
            pt[chunk][pp] = st[j];
        }
    }
#else
    if (!half) {
        pm[chunk][pix] = sm[0];
        ps[chunk][pix] = ss[0];
        pt[chunk][pix] = st[0];
    }
#endif
    __syncthreads();

    // ---- merge the 4 chunk partials, emit expected depth ----
    if (tid < PIX_PER_BLOCK) {
        float M = pm[0][tid];
#pragma unroll
        for (int c = 1; c < CHUNKS; ++c) M = fmaxf(M, pm[c][tid]);
        float S = 0.f, T = 0.f;
#pragma unroll
        for (int c = 0; c < CHUNKS; ++c) {
            const float e = __expf(pm[c][tid] - M);
            S = fmaf(ps[c][tid], e, S);
            T = fmaf(pt[c][tid], e, T);
        }
        out[blockIdx.x * PIX_PER_BLOCK + tid] = T / S;
    }
}

extern "C" void kernel_launch(void* const* d_in, const int* in_sizes, int n_in,
                              void* d_out, int out_size, void* d_ws, size_t ws_size,
                              hipStream_t stream) {
    const float* feat  = (const float*)d_in[0];   // features [V,B,C,H,W]
    const float* proj  = (const float*)d_in[1];   // proj_matrices [B,V,3,4]
    const float* dvals = (const float*)d_in[2];   // depth_values [B,D,H,W]
    const float* regw  = (const float*)d_in[3];   // reg_weight [C]
    float* out = (float*)d_out;                   // depth [B,H,W]
    (void)in_sizes; (void)n_in; (void)out_size; (void)d_ws; (void)ws_size;

    const int nblocks = (B_ * H_ * W_) / PIX_PER_BLOCK;   // 1280
    hipLaunchKernelGGL(mvs_depth_kernel, dim3(nblocks), dim3(256), 0, stream,
                       feat, proj, dvals, regw, out);
}